// ConditionalDLFactorized18_74680891343528
// MI455X (gfx1250) — compile-verified
//
#include <hip/hip_runtime.h>
#include <hip/hip_bf16.h>

// Problem constants (from reference): T=512, B=16 -> 8192 tokens, C=512,
// REDUCE=64, OUT=512, NBITS=8 -> 256 experts.
#define NTOK   8192
#define CDIM   512
#define RDIM   64
#define ODIM   512
#define NEXP   256
#define WROW   (ODIM * RDIM)           // 32768 floats per expert row
#define BUCKET_CAP (NTOK + NEXP * 16)  // 16-aligned bucket bases need padding

typedef float v2f __attribute__((ext_vector_type(2)));
typedef float v8f __attribute__((ext_vector_type(8)));
typedef unsigned int uint32;

// ---------------------------------------------------------------------------
// Kernel 0: zero histogram, zero loss output
// ---------------------------------------------------------------------------
__global__ void k0_init(int* __restrict__ counts, float* __restrict__ out) {
    if (threadIdx.x < NEXP) counts[threadIdx.x] = 0;
    if (threadIdx.x == 0) out[(size_t)NTOK * ODIM] = 0.0f;  // loss scalar
}

// ---------------------------------------------------------------------------
// Kernel 1: per token, compute v[r] = pw_w1[r,:] . x  and the 8 sign bits of
// map_w @ x + map_b  ->  qz1 code; histogram codes.
// One block (128 threads) per token; x row staged in LDS.
// (saturated sigmoid > 0.5  <=>  pre-activation k > 0)
// ---------------------------------------------------------------------------
__device__ __forceinline__ float dot512(const float* __restrict__ wrow,
                                        const float* __restrict__ xs) {
    const float4* w4 = (const float4*)wrow;
    const float4* x4 = (const float4*)xs;
    float acc = 0.0f;
#pragma unroll 8
    for (int i = 0; i < CDIM / 4; ++i) {
        float4 w = w4[i], xv = x4[i];
        acc = fmaf(w.x, xv.x, acc);
        acc = fmaf(w.y, xv.y, acc);
        acc = fmaf(w.z, xv.z, acc);
        acc = fmaf(w.w, xv.w, acc);
    }
    return acc;
}

__global__ __launch_bounds__(128) void k1_embed(
    const float* __restrict__ x, const float* __restrict__ map_w,
    const float* __restrict__ map_b, const float* __restrict__ pw_w1,
    float* __restrict__ v, int* __restrict__ qz, int* __restrict__ counts) {
    __shared__ float xs[CDIM];
    __shared__ int code_s;
    const int tok = blockIdx.x;
    const int tid = threadIdx.x;

    ((float4*)xs)[tid] = ((const float4*)(x + (size_t)tok * CDIM))[tid];
    if (tid == 0) code_s = 0;
    __syncthreads();

    if (tid < RDIM) {
        v[(size_t)tok * RDIM + tid] = dot512(pw_w1 + (size_t)tid * CDIM, xs);
    } else if (tid < RDIM + 8) {
        const int j = tid - RDIM;
        float k = dot512(map_w + (size_t)j * CDIM, xs) + map_b[j];
        if (k > 0.0f) atomicOr(&code_s, 1 << j);
    }
    __syncthreads();
    if (tid == 0) {
        qz[tok] = code_s;
        atomicAdd(&counts[code_s], 1);
    }
}

// ---------------------------------------------------------------------------
// Kernel 2: exclusive scan over 256 expert counts; bucket bases rounded up to
// a multiple of 16 so every 16-token chunk is 16-aligned in the bucket array
// (enables b128 token-id loads in the WMMA kernel).
// ---------------------------------------------------------------------------
__global__ void k2_scan(const int* __restrict__ counts,
                        int* __restrict__ basearr, int* __restrict__ cursor) {
    const int e = threadIdx.x;
    int b = 0;
    for (int i = 0; i < e; ++i) b += (counts[i] + 15) & ~15;
    basearr[e] = b;
    cursor[e] = b;
}

// ---------------------------------------------------------------------------
// Kernel 3: scatter token ids into expert buckets
// ---------------------------------------------------------------------------
__global__ void k3_scatter(const int* __restrict__ qz, int* __restrict__ cursor,
                           int* __restrict__ bucket) {
    const int t = blockIdx.x * blockDim.x + threadIdx.x;
    if (t < NTOK) {
        const int c = qz[t];
        const int pos = atomicAdd(&cursor[c], 1);
        bucket[pos] = t;
    }
}

// ---------------------------------------------------------------------------
// Kernel 4: per (expert e, 16-wide out tile), one wave.
// B fragments = (w21[e] + w22[255-e]) tile, built once, reused for every
// 16-token chunk of the bucket. D = A(16 tok x 4) x B(4 x 16 out) + C via
// V_WMMA_F32_16X16X4_F32, K=64 in 16 steps. Full fp32 path.
//
// f32 A layout (16x4): lanes 0-15 -> M=lane, VGPR0/1 = K0/K1;
//                      lanes 16-31 -> M=lane-16, VGPR0/1 = K2/K3.
// C/D layout: VGPR j -> M = j + 8*(lane>=16), N = lane%16.
// Bias is folded into the C operand of the first WMMA (all elements a lane
// holds share N = orow). Stores index y with 32-bit offsets so the backend
// can use saddr + voffset (scale_offset) addressing.
// ---------------------------------------------------------------------------
__device__ __forceinline__ v8f wmma_chunk16(const float* __restrict__ vr,
                                            const v2f* __restrict__ bfrag,
                                            int kbase, float bv) {
    v8f acc;
#pragma unroll
    for (int j = 0; j < 8; ++j) acc[j] = bv;   // bias pre-loaded into C
#pragma unroll
    for (int c = 0; c < 16; ++c) {
        v2f a;
        a.x = vr[4 * c + kbase];
        a.y = vr[4 * c + kbase + 1];
        acc = __builtin_amdgcn_wmma_f32_16x16x4_f32(
            false, a, false, bfrag[c], (short)0, acc, false, false);
    }
    return acc;
}

__global__ __launch_bounds__(32) void k4_wmma(
    const float* __restrict__ v, const int* __restrict__ bucket,
    const int* __restrict__ basearr, const int* __restrict__ counts,
    const float* __restrict__ w21, const float* __restrict__ w22,
    const float* __restrict__ bias, float* __restrict__ y) {
    const int e = blockIdx.y;
    const int cnt = counts[e];
    if (cnt == 0) return;
    const int base = basearr[e];           // multiple of 16

    const int lane = threadIdx.x;
    const int nloc = lane & 15;
    const int half = lane >> 4;            // 0: K{0,1}/M 0-7 ; 1: K{2,3}/M 8-15
    const int kbase = half * 2;
    const int orow = blockIdx.x * 16 + nloc;

    // Summed expert weight tile -> B fragments (register resident)
    const float* r21 = w21 + (size_t)e * WROW + (size_t)orow * RDIM;
    const float* r22 = w22 + (size_t)(NEXP - 1 - e) * WROW + (size_t)orow * RDIM;
    v2f bfrag[16];
#pragma unroll
    for (int c = 0; c < 16; ++c) {
        const int k = 4 * c + kbase;
        bfrag[c].x = r21[k] + r22[k];
        bfrag[c].y = r21[k + 1] + r22[k + 1];
    }
    const float bv = bias[orow];
    const uint32 ob = (uint32)orow;        // 32-bit store offset base

    const int nfull = cnt & ~15;

    // ---- full 16-token chunks: no clamps, no masked stores ----
    for (int i0 = 0; i0 < nfull; i0 += 16) {
        const float* vr =
            v + ((uint32)bucket[base + i0 + nloc] * (uint32)RDIM);
        v8f acc = wmma_chunk16(vr, bfrag, kbase, bv);

        // token ids for this lane's 8 D rows: contiguous, 32B-aligned
        const int4* bp = (const int4*)(bucket + base + i0 + half * 8);
        const int4 t0 = bp[0];
        const int4 t1 = bp[1];
        y[(uint32)t0.x * (uint32)ODIM + ob] = acc[0];
        y[(uint32)t0.y * (uint32)ODIM + ob] = acc[1];
        y[(uint32)t0.z * (uint32)ODIM + ob] = acc[2];
        y[(uint32)t0.w * (uint32)ODIM + ob] = acc[3];
        y[(uint32)t1.x * (uint32)ODIM + ob] = acc[4];
        y[(uint32)t1.y * (uint32)ODIM + ob] = acc[5];
        y[(uint32)t1.z * (uint32)ODIM + ob] = acc[6];
        y[(uint32)t1.w * (uint32)ODIM + ob] = acc[7];
    }

    // ---- single ragged tail chunk ----
    if (nfull < cnt) {
        int mi = nfull + nloc;
        if (mi >= cnt) mi = cnt - 1;       // clamp (loads only; EXEC stays full)
        const float* vr = v + ((uint32)bucket[base + mi] * (uint32)RDIM);
        v8f acc = wmma_chunk16(vr, bfrag, kbase, bv);

#pragma unroll
        for (int j = 0; j < 8; ++j) {
            const int m = j + half * 8;
            if (nfull + m < cnt) {
                const uint32 tok = (uint32)bucket[base + nfull + m];
                y[tok * (uint32)ODIM + ob] = acc[j];
            }
        }
    }
}

// ---------------------------------------------------------------------------
extern "C" void kernel_launch(void* const* d_in, const int* in_sizes, int n_in,
                              void* d_out, int out_size, void* d_ws,
                              size_t ws_size, hipStream_t stream) {
    const float* x      = (const float*)d_in[0];
    // d_in[1] ("key") is overwritten by x in the reference -> unused
    const float* map_w  = (const float*)d_in[2];
    const float* map_b  = (const float*)d_in[3];
    const float* pw_w1  = (const float*)d_in[4];
    const float* pw_w21 = (const float*)d_in[5];
    const float* pw_w22 = (const float*)d_in[6];
    const float* pw_B   = (const float*)d_in[7];
    float* y = (float*)d_out;

    // Workspace layout
    float* v      = (float*)d_ws;                 // NTOK * RDIM floats (2 MB)
    int* qz       = (int*)(v + (size_t)NTOK * RDIM);
    int* bucket   = qz + NTOK;                    // BUCKET_CAP entries
    int* counts   = bucket + BUCKET_CAP;
    int* cursor   = counts + NEXP;
    int* basearr  = cursor + NEXP;

    k0_init<<<1, 256, 0, stream>>>(counts, y);
    k1_embed<<<NTOK, 128, 0, stream>>>(x, map_w, map_b, pw_w1, v, qz, counts);
    k2_scan<<<1, NEXP, 0, stream>>>(counts, basearr, cursor);
    k3_scatter<<<NTOK / 256, 256, 0, stream>>>(qz, cursor, bucket);
    dim3 grid(ODIM / 16, NEXP);
    k4_wmma<<<grid, 32, 0, stream>>>(v, bucket, basearr, counts, pw_w21,
                                     pw_w22, pw_B, y);
}